// GNN_19353122636479
// MI455X (gfx1250) — compile-verified
//
#include <hip/hip_runtime.h>
#include <hip/hip_bf16.h>

// ---------------------------------------------------------------------------
// GNN (3x GINEConv + pooling head) for MI455X / gfx1250, wave32 + WMMA fp32.
// Edge-embedding chain and node MLPs run on V_WMMA_F32_16X16X4_F32 (full fp32).
// Scatter-add via hardware global_atomic_add_f32 (agg fits in 192MB L2).
// ---------------------------------------------------------------------------

typedef __attribute__((ext_vector_type(2))) float v2f;
typedef __attribute__((ext_vector_type(8))) float v8f;

__device__ __forceinline__ v8f vz8() {
    v8f z = {0.f, 0.f, 0.f, 0.f, 0.f, 0.f, 0.f, 0.f};
    return z;
}

__device__ __forceinline__ v8f wmma4(v2f a, v2f b, v8f c) {
    // D = A(16x4,f32) * B(4x16,f32) + C(16x16,f32)
    return __builtin_amdgcn_wmma_f32_16x16x4_f32(false, a, false, b, (short)0, c,
                                                 false, false);
}

// C[16x64] += A[16xK] @ W[Kx64], A in LDS (row-major, stride lda), W row-major.
// A frag layout: lanes 0-15 hold K=k,k+1 ; lanes 16-31 hold K=k+2,k+3 (row M=lane%16)
// B frag mirrors; C/D: VGPR v -> row v (lanes 0-15) / v+8 (lanes 16-31), col lane%16.
template <int K>
__device__ __forceinline__ void wave_gemm64(const float* As, int lda,
                                            const float* Ws, v8f acc[4], int lane) {
    const int m  = lane & 15;
    const int kh = (lane & 16) ? 2 : 0;
#pragma unroll
    for (int n = 0; n < 4; ++n) {
        v8f c = acc[n];
#pragma unroll
        for (int kk = 0; kk < K; kk += 4) {
            v2f a, b;
            a.x = As[m * lda + kk + kh];
            a.y = As[m * lda + kk + kh + 1];
            b.x = Ws[(kk + kh) * 64 + n * 16 + m];
            b.y = Ws[(kk + kh + 1) * 64 + n * 16 + m];
            c = wmma4(a, b, c);
        }
        acc[n] = c;
    }
}

// Single 16x16 output tile, K=4 (for ea1 = edge_attr[16x4] @ ee1_w[4x16]).
__device__ __forceinline__ v8f wave_gemm16_k4(const float* As, int lda,
                                              const float* Ws, int ldw, int lane) {
    const int m  = lane & 15;
    const int kh = (lane & 16) ? 2 : 0;
    v2f a, b;
    a.x = As[m * lda + kh];
    a.y = As[m * lda + kh + 1];
    b.x = Ws[kh * ldw + m];
    b.y = Ws[(kh + 1) * ldw + m];
    return wmma4(a, b, vz8());
}

// ---------------------------------------------------------------------------
// Message kernels: per 16-edge tile, recompute ea chain with WMMA, then
// msg = relu(h[src] + eaL); atomicAdd into agg[dst].
// ---------------------------------------------------------------------------
template <int LAYER>
__global__ __launch_bounds__(64) void msg_kernel(
    const float* __restrict__ edge_attr, const int* __restrict__ edge_index,
    const float* __restrict__ hprev,
    const float* __restrict__ ee1w, const float* __restrict__ ee1b,
    const float* __restrict__ ee2w, const float* __restrict__ ee2b,
    const float* __restrict__ ee3w, const float* __restrict__ ee3b,
    float* __restrict__ agg, int E) {
    constexpr int WPB = 2;  // waves per block
    __shared__ float sW1[4 * 16];
    __shared__ float sB1[16];
    __shared__ float sW2[(LAYER >= 2) ? 16 * 64 : 1];
    __shared__ float sB2[(LAYER >= 2) ? 64 : 1];
    __shared__ float sW3[(LAYER == 3) ? 64 * 64 : 1];
    __shared__ float sB3[(LAYER == 3) ? 64 : 1];
    __shared__ float sEA[WPB][16 * 9];                       // 16x4 edge_attr (pad 9)
    __shared__ float sT1[WPB][16 * 20];                      // ea1 16x16 (pad 20)
    __shared__ float sT2[(LAYER >= 2) ? WPB : 1][16 * 68];   // ea2 16x64 (pad 68)
    __shared__ float sT3[(LAYER == 3) ? WPB : 1][16 * 68];   // ea3 16x64

    const int tid = threadIdx.x;
    const int wv = tid >> 5, lane = tid & 31;

    for (int i = tid; i < 64; i += 64) sW1[i] = ee1w[i];
    if (tid < 16) sB1[tid] = ee1b[tid];
    if (LAYER >= 2) {
        for (int i = tid; i < 16 * 64; i += 64) sW2[i] = ee2w[i];
        if (tid < 64) sB2[tid] = ee2b[tid];
    }
    if (LAYER == 3) {
        for (int i = tid; i < 64 * 64; i += 64) sW3[i] = ee3w[i];
        if (tid < 64) sB3[tid] = ee3b[tid];
    }
    __syncthreads();

    const int ebase = (blockIdx.x * WPB + wv) * 16;

    // Load 16x4 edge_attr tile (one float4 per row).
    if (lane < 16) {
        int e = ebase + lane;
        if (e >= E) e = E - 1;
        const float4 v = *(const float4*)(edge_attr + (size_t)e * 4);
        float* d = &sEA[wv][lane * 9];
        d[0] = v.x; d[1] = v.y; d[2] = v.z; d[3] = v.w;
    }

    // ea1 = edge_attr @ ee1_w + b  (single WMMA)
    {
        v8f c = wave_gemm16_k4(sEA[wv], 9, sW1, 16, lane);
        const int m = lane & 15, rb = (lane & 16) ? 8 : 0;
#pragma unroll
        for (int v = 0; v < 8; ++v) sT1[wv][(rb + v) * 20 + m] = c[v] + sB1[m];
    }
    if (LAYER >= 2) {  // ea2 = ea1 @ ee2_w + b  (16 WMMA)
        v8f acc[4] = {vz8(), vz8(), vz8(), vz8()};
        wave_gemm64<16>(sT1[wv], 20, sW2, acc, lane);
        const int m = lane & 15, rb = (lane & 16) ? 8 : 0;
#pragma unroll
        for (int n = 0; n < 4; ++n)
#pragma unroll
            for (int v = 0; v < 8; ++v)
                sT2[wv][(rb + v) * 68 + n * 16 + m] = acc[n][v] + sB2[n * 16 + m];
    }
    if (LAYER == 3) {  // ea3 = ea2 @ ee3_w + b  (64 WMMA)
        v8f acc[4] = {vz8(), vz8(), vz8(), vz8()};
        wave_gemm64<64>(sT2[wv], 68, sW3, acc, lane);
        const int m = lane & 15, rb = (lane & 16) ? 8 : 0;
#pragma unroll
        for (int n = 0; n < 4; ++n)
#pragma unroll
            for (int v = 0; v < 8; ++v)
                sT3[wv][(rb + v) * 68 + n * 16 + m] = acc[n][v] + sB3[n * 16 + m];
    }

    // Gather h[src], relu(h+ea), scatter-add to agg[dst].  2 lanes per edge row.
    const float* ea = (LAYER == 1) ? sT1[wv] : ((LAYER == 2) ? sT2[wv] : sT3[wv]);
    const int eald = (LAYER == 1) ? 20 : 68;
    constexpr int D = (LAYER == 1) ? 16 : 64;
    const int r = lane >> 1, p = lane & 1;
    int e = ebase + r;
    const bool ok = (e < E);
    if (!ok) e = E - 1;
    const int sn = edge_index[e];
    const int dn = edge_index[(size_t)E + e];
    const float* hs = hprev + (size_t)sn * D;
    float* ag = agg + (size_t)dn * D;
    const int c0 = p * (D / 2);
#pragma unroll
    for (int c = 0; c < D / 2; c += 4) {
        const float4 hv = *(const float4*)(hs + c0 + c);
        const float* ep = &ea[r * eald + c0 + c];
        const float m0 = fmaxf(hv.x + ep[0], 0.f);
        const float m1 = fmaxf(hv.y + ep[1], 0.f);
        const float m2 = fmaxf(hv.z + ep[2], 0.f);
        const float m3 = fmaxf(hv.w + ep[3], 0.f);
        if (ok) {
            unsafeAtomicAdd(ag + c0 + c + 0, m0);
            unsafeAtomicAdd(ag + c0 + c + 1, m1);
            unsafeAtomicAdd(ag + c0 + c + 2, m2);
            unsafeAtomicAdd(ag + c0 + c + 3, m3);
        }
    }
}

// ---------------------------------------------------------------------------
// Node MLP kernels: h_out = relu(LN(relu((hin+agg)@W1+b1)@W2+b2 [+ hin]))
// ---------------------------------------------------------------------------
template <int IN, int RESID>
__global__ __launch_bounds__(64) void mlp_kernel(
    const float* __restrict__ hin, const float* __restrict__ agg, int aggld,
    const float* __restrict__ w1, const float* __restrict__ b1,
    const float* __restrict__ w2, const float* __restrict__ b2,
    const float* __restrict__ lng, const float* __restrict__ lnb,
    float* __restrict__ hout, int N) {
    constexpr int WPB = 2;
    __shared__ float sW1[IN * 64];
    __shared__ float sW2[64 * 64];
    __shared__ float sB1v[64], sB2v[64], sG[64], sBv[64];
    __shared__ float sA[WPB][16 * 68];
    __shared__ float sT[WPB][16 * 68];

    const int tid = threadIdx.x, wv = tid >> 5, lane = tid & 31;
    for (int i = tid; i < IN * 64; i += 64) sW1[i] = w1[i];
    for (int i = tid; i < 64 * 64; i += 64) sW2[i] = w2[i];
    if (tid < 64) {
        sB1v[tid] = b1[tid]; sB2v[tid] = b2[tid];
        sG[tid] = lng[tid];  sBv[tid] = lnb[tid];
    }
    __syncthreads();

    const int nbase = (blockIdx.x * WPB + wv) * 16;
    const int r = lane >> 1, p = lane & 1;
    int node = nbase + r;
    if (node >= N) node = N - 1;

    // A = hin + agg  (tile 16 x IN)
    {
        const float* hp = hin + (size_t)node * IN;
        const float* ap = agg + (size_t)node * aggld;
        const int c0 = p * (IN / 2);
#pragma unroll
        for (int c = 0; c < IN / 2; c += 4) {
            const float4 hv = *(const float4*)(hp + c0 + c);
            const float4 av = *(const float4*)(ap + c0 + c);
            float* d = &sA[wv][r * 68 + c0 + c];
            d[0] = hv.x + av.x; d[1] = hv.y + av.y;
            d[2] = hv.z + av.z; d[3] = hv.w + av.w;
        }
    }
    // GEMM1 + bias + relu -> sT
    {
        v8f acc[4] = {vz8(), vz8(), vz8(), vz8()};
        wave_gemm64<IN>(sA[wv], 68, sW1, acc, lane);
        const int m = lane & 15, rb = (lane & 16) ? 8 : 0;
#pragma unroll
        for (int n = 0; n < 4; ++n)
#pragma unroll
            for (int v = 0; v < 8; ++v)
                sT[wv][(rb + v) * 68 + n * 16 + m] =
                    fmaxf(acc[n][v] + sB1v[n * 16 + m], 0.f);
    }
    // GEMM2 + bias (+ residual) -> sA (reuse)
    {
        v8f acc[4] = {vz8(), vz8(), vz8(), vz8()};
        wave_gemm64<64>(sT[wv], 68, sW2, acc, lane);
        const int m = lane & 15, rb = (lane & 16) ? 8 : 0;
#pragma unroll
        for (int n = 0; n < 4; ++n)
#pragma unroll
            for (int v = 0; v < 8; ++v) {
                float val = acc[n][v] + sB2v[n * 16 + m];
                if (RESID) {
                    int nd = nbase + rb + v;
                    if (nd >= N) nd = N - 1;
                    val += hin[(size_t)nd * IN + n * 16 + m];
                }
                sA[wv][(rb + v) * 68 + n * 16 + m] = val;
            }
    }
    // Row LayerNorm (64) + relu + store.  2 lanes per row, shfl_xor combine.
    {
        const int c0 = p * 32;
        float4 keep[8];
        float s = 0.f, q = 0.f;
#pragma unroll
        for (int c = 0; c < 32; c += 4) {
            const float4 t = *(const float4*)(&sA[wv][r * 68 + c0 + c]);
            keep[c / 4] = t;
            s += t.x + t.y + t.z + t.w;
            q += t.x * t.x + t.y * t.y + t.z * t.z + t.w * t.w;
        }
        s += __shfl_xor(s, 1, 32);
        q += __shfl_xor(q, 1, 32);
        const float mean = s * (1.f / 64.f);
        const float var = q * (1.f / 64.f) - mean * mean;
        const float rstd = rsqrtf(var + 1e-5f);
        const int nd = nbase + r;
        if (nd < N) {
            float* op = hout + (size_t)nd * 64 + c0;
#pragma unroll
            for (int c = 0; c < 32; c += 4) {
                const float4 t = keep[c / 4];
                float4 o;
                o.x = fmaxf((t.x - mean) * rstd * sG[c0 + c + 0] + sBv[c0 + c + 0], 0.f);
                o.y = fmaxf((t.y - mean) * rstd * sG[c0 + c + 1] + sBv[c0 + c + 1], 0.f);
                o.z = fmaxf((t.z - mean) * rstd * sG[c0 + c + 2] + sBv[c0 + c + 2], 0.f);
                o.w = fmaxf((t.w - mean) * rstd * sG[c0 + c + 3] + sBv[c0 + c + 3], 0.f);
                *(float4*)(op + c) = o;
            }
        }
    }
}

// ---------------------------------------------------------------------------
// Per-graph pooling: sum + count (f32 atomics) and max via int-bit atomicMax
// (valid: h >= 0 after relu; init 0 reproduces where(cnt>0, mx, 0)).
// ---------------------------------------------------------------------------
__global__ __launch_bounds__(256) void pool_kernel(
    const float* __restrict__ h3, const int* __restrict__ batch,
    float* __restrict__ ps, float* __restrict__ pc, float* __restrict__ pmx, int N) {
    const int idx = blockIdx.x * 256 + threadIdx.x;
    if (idx >= N * 64) return;
    const int n = idx >> 6, c = idx & 63;
    const int g = batch[n];
    const float v = h3[idx];
    unsafeAtomicAdd(ps + (size_t)g * 64 + c, v);
    atomicMax((int*)pmx + (size_t)g * 64 + c, __float_as_int(v));
    if (c == 0) unsafeAtomicAdd(pc + g, 1.0f);
}

// ---------------------------------------------------------------------------
// Graph head: [sum|mean|max](192) @ f_w1 -> LN -> relu -> @ f_w2 + b.
// One 64-thread block per graph.
// ---------------------------------------------------------------------------
__global__ __launch_bounds__(64) void head_kernel(
    const float* __restrict__ ps, const float* __restrict__ pc,
    const float* __restrict__ pmx,
    const float* __restrict__ fw1, const float* __restrict__ fb1,
    const float* __restrict__ flg, const float* __restrict__ flb,
    const float* __restrict__ fw2, const float* __restrict__ fb2,
    float* __restrict__ out, int G) {
    const int g = blockIdx.x, t = threadIdx.x;
    __shared__ float xg[192];
    __shared__ float rs[64], rq[64];
    const float cnt = pc[g];
    const float s = ps[(size_t)g * 64 + t];
    xg[t] = s;
    xg[64 + t] = s / fmaxf(cnt, 1.0f);
    xg[128 + t] = pmx[(size_t)g * 64 + t];
    __syncthreads();

    float a = fb1[t];
#pragma unroll 4
    for (int k = 0; k < 192; ++k) a += xg[k] * fw1[k * 64 + t];

    rs[t] = a;
    rq[t] = a * a;
    __syncthreads();
    for (int o = 32; o > 0; o >>= 1) {
        if (t < o) { rs[t] += rs[t + o]; rq[t] += rq[t + o]; }
        __syncthreads();
    }
    const float mean = rs[0] * (1.f / 64.f);
    const float var = rq[0] * (1.f / 64.f) - mean * mean;
    const float rstd = rsqrtf(var + 1e-5f);
    __syncthreads();
    const float hn = fmaxf((a - mean) * rstd * flg[t] + flb[t], 0.f);
    rs[t] = hn * fw2[t];
    __syncthreads();
    for (int o = 32; o > 0; o >>= 1) {
        if (t < o) rs[t] += rs[t + o];
        __syncthreads();
    }
    if (t == 0) out[g] = rs[0] + fb2[0];
}

// ---------------------------------------------------------------------------
extern "C" void kernel_launch(void* const* d_in, const int* in_sizes, int n_in,
                              void* d_out, int out_size, void* d_ws, size_t ws_size,
                              hipStream_t stream) {
    const float* x  = (const float*)d_in[0];
    const float* ea = (const float*)d_in[1];
    const int* ei   = (const int*)d_in[2];
    const int* batch = (const int*)d_in[3];
    const float* ee1w = (const float*)d_in[4];  const float* ee1b = (const float*)d_in[5];
    const float* ee2w = (const float*)d_in[6];  const float* ee2b = (const float*)d_in[7];
    const float* ee3w = (const float*)d_in[8];  const float* ee3b = (const float*)d_in[9];
    const float* m1w1 = (const float*)d_in[10]; const float* m1b1 = (const float*)d_in[11];
    const float* m1w2 = (const float*)d_in[12]; const float* m1b2 = (const float*)d_in[13];
    const float* m2w1 = (const float*)d_in[14]; const float* m2b1 = (const float*)d_in[15];
    const float* m2w2 = (const float*)d_in[16]; const float* m2b2 = (const float*)d_in[17];
    const float* m3w1 = (const float*)d_in[18]; const float* m3b1 = (const float*)d_in[19];
    const float* m3w2 = (const float*)d_in[20]; const float* m3b2 = (const float*)d_in[21];
    const float* ln1g = (const float*)d_in[22]; const float* ln1b = (const float*)d_in[23];
    const float* ln2g = (const float*)d_in[24]; const float* ln2b = (const float*)d_in[25];
    const float* ln3g = (const float*)d_in[26]; const float* ln3b = (const float*)d_in[27];
    const float* fw1  = (const float*)d_in[28]; const float* fb1  = (const float*)d_in[29];
    const float* flg  = (const float*)d_in[30]; const float* flb  = (const float*)d_in[31];
    const float* fw2  = (const float*)d_in[32]; const float* fb2  = (const float*)d_in[33];

    const int N = in_sizes[0] / 16;
    const int E = in_sizes[1] / 4;
    const int G = out_size;
    const size_t N64 = (size_t)N * 64;

    float* ws  = (float*)d_ws;
    float* h1  = ws;
    float* h2  = h1 + N64;
    float* h3  = h2 + N64;
    float* agg = h3 + N64;
    float* ps  = agg + N64;
    float* pc  = ps + (size_t)G * 64;
    float* pmx = pc + G;

    const int msgBlocks = (E + 31) / 32;   // 2 waves x 16 edges per block
    const int mlpBlocks = (N + 31) / 32;   // 2 waves x 16 nodes per block

    // ---- layer 1 (D=16) ----
    hipMemsetAsync(agg, 0, (size_t)N * 16 * sizeof(float), stream);
    msg_kernel<1><<<msgBlocks, 64, 0, stream>>>(ea, ei, x, ee1w, ee1b, ee2w, ee2b,
                                                ee3w, ee3b, agg, E);
    mlp_kernel<16, 0><<<mlpBlocks, 64, 0, stream>>>(x, agg, 16, m1w1, m1b1, m1w2,
                                                    m1b2, ln1g, ln1b, h1, N);
    // ---- layer 2 (D=64) ----
    hipMemsetAsync(agg, 0, N64 * sizeof(float), stream);
    msg_kernel<2><<<msgBlocks, 64, 0, stream>>>(ea, ei, h1, ee1w, ee1b, ee2w, ee2b,
                                                ee3w, ee3b, agg, E);
    mlp_kernel<64, 0><<<mlpBlocks, 64, 0, stream>>>(h1, agg, 64, m2w1, m2b1, m2w2,
                                                    m2b2, ln2g, ln2b, h2, N);
    // ---- layer 3 (D=64, residual) ----
    hipMemsetAsync(agg, 0, N64 * sizeof(float), stream);
    msg_kernel<3><<<msgBlocks, 64, 0, stream>>>(ea, ei, h2, ee1w, ee1b, ee2w, ee2b,
                                                ee3w, ee3b, agg, E);
    mlp_kernel<64, 1><<<mlpBlocks, 64, 0, stream>>>(h2, agg, 64, m3w1, m3b1, m3w2,
                                                    m3b2, ln3g, ln3b, h3, N);
    // ---- pooling + head ----
    hipMemsetAsync(ps, 0, ((size_t)G * 129) * sizeof(float), stream);
    pool_kernel<<<(N * 64 + 255) / 256, 256, 0, stream>>>(h3, batch, ps, pc, pmx, N);
    head_kernel<<<G, 64, 0, stream>>>(ps, pc, pmx, fw1, fb1, flg, flb, fw2, fb2,
                                      (float*)d_out, G);
}